// DYSProjector_53652731461872
// MI455X (gfx1250) — compile-verified
//
#include <hip/hip_runtime.h>

typedef __attribute__((ext_vector_type(2))) float v2f;
typedef __attribute__((ext_vector_type(8))) float v8f;
typedef __attribute__((ext_vector_type(4))) unsigned int u32x4;
typedef __attribute__((ext_vector_type(8))) int i32x8;
typedef __attribute__((ext_vector_type(4))) int i32x4;

#define DYS_ALPHA 0.5f
#define DYS_TOL   1e-2f

// Single-wave workgroup: LDS ops complete in order within a wave (ISA 7.3),
// so a compiler scheduling fence is sufficient for cross-lane LDS traffic.
__device__ __forceinline__ void wave_sync() {
    __asm__ __volatile__("" ::: "memory");
    __builtin_amdgcn_wave_barrier();
    __asm__ __volatile__("" ::: "memory");
}

// One wave32 per batch element. m=32, n=16, N=64.
__global__ __launch_bounds__(32)
void dys_projector_kernel(const float* __restrict__ u_nom,   // (B,16)
                          const float* __restrict__ A,       // (B,32,16)
                          const float* __restrict__ b,       // (B,32,1)
                          const int*   __restrict__ max_iter_p,
                          float* __restrict__ out_u,         // (B,16)
                          float* __restrict__ out_z)         // (B,64)
{
    constexpr int m = 32, n = 16;
    const int bidx = blockIdx.x;
    const int lane = threadIdx.x;          // 0..31, wave32

    __shared__ __align__(16) float Alds[m * n];   // A, row-major 32x16
    __shared__ __align__(16) float GJ[m * 64];    // [G | I] -> [I | G^-1]
    __shared__ __align__(16) float bcd[16];       // broadcast d
    __shared__ __align__(16) float bcr[32];       // broadcast r
    __shared__ __align__(16) float bcw[32];       // broadcast w

    const float* Ab = A + (size_t)bidx * (m * n);

    // ---- TDM: async DMA A tile (512 f32, 2KB) global -> LDS ----
    {
        unsigned long long ga = (unsigned long long)(uintptr_t)Ab;
        unsigned lds0 = (unsigned)(uintptr_t)(void*)Alds;   // low 32b = LDS offset

        u32x4 g0;
        g0[0] = 1u;                                   // count=1 (valid user D#)
        g0[1] = lds0;                                 // lds_addr
        g0[2] = (unsigned)(ga & 0xffffffffu);         // global_addr[31:0]
        g0[3] = (unsigned)((ga >> 32) & 0x01ffffffu)  // global_addr[56:32]
              | 0x80000000u;                          // type=2 ("image")

        i32x8 g1;
        g1[0] = (int)(2u << 16);       // workgroup_mask=0 (no cluster), data_size=4B
        g1[1] = (int)(512u << 16);     // tensor_dim0 = 512 (low 16 bits)
        g1[2] = (int)(1u << 16);       // tensor_dim0 hi=0, tensor_dim1 = 1
        g1[3] = (int)(512u << 16);     // tensor_dim1 hi=0, tile_dim0 = 512
        g1[4] = 1;                     // tile_dim1 = 1, tile_dim2 = 0
        g1[5] = 512;                   // tensor_dim0_stride[31:0]
        g1[6] = 0;                     // stride hi, tensor_dim1_stride lo
        g1[7] = 0;

        i32x4 gz  = {0, 0, 0, 0};                 // groups 2/3 unused (<=2D)
        i32x8 gz8 = {0, 0, 0, 0, 0, 0, 0, 0};     // extra group (clang-23 form)
#if defined(__gfx1250__) && __has_builtin(__builtin_amdgcn_tensor_load_to_lds)
        // amdgpu-toolchain (clang-23) 6-arg form:
        // (u32x4 g0, i32x8 g1, i32x4 g2, i32x4 g3, i32x8 extra, i32 cpol)
        __builtin_amdgcn_tensor_load_to_lds(g0, g1, gz, gz, gz8, 0);
        __builtin_amdgcn_s_wait_tensorcnt(0);
#else
        (void)g0; (void)g1; (void)gz; (void)gz8;
        #pragma unroll
        for (int j = 0; j < n; j += 4)
            *(float4*)&Alds[lane * n + j] = *(const float4*)(Ab + lane * n + j);
#endif
    }
    wave_sync();

    __builtin_prefetch(b + (size_t)bidx * 32, 0, 1);   // global_prefetch_b8

    // lane l keeps row l of A, and column (l&15) of A, in VGPRs
    float Ar[n];
    #pragma unroll
    for (int j = 0; j < n; j += 4) {
        float4 v4 = *(const float4*)&Alds[lane * n + j];
        Ar[j] = v4.x; Ar[j+1] = v4.y; Ar[j+2] = v4.z; Ar[j+3] = v4.w;
    }
    float Atcol[m];
    #pragma unroll
    for (int i = 0; i < m; ++i) Atcol[i] = Alds[i * n + (lane & 15)];

    // ---- G = 2*A*A^T + I via V_WMMA_F32_16X16X4_F32 ----
    // A-frag (16x4 f32): lanes 0-15 -> K={0,1}, lanes 16-31 -> K={2,3}, M = lane&15.
    // For A*A^T the B-frag of column-tile bj reads the same LDS pattern as the
    // A-frag of row-tile bj, so fragments are shared.
    const int mrow = lane & 15;
    const int ksel = (lane < 16) ? 0 : 2;
    v8f acc00 = {}, acc01 = {}, acc10 = {}, acc11 = {};
    #pragma unroll
    for (int kb = 0; kb < 4; ++kb) {
        v2f f0, f1;
        f0.x = Alds[( 0 + mrow) * n + kb * 4 + ksel + 0];
        f0.y = Alds[( 0 + mrow) * n + kb * 4 + ksel + 1];
        f1.x = Alds[(16 + mrow) * n + kb * 4 + ksel + 0];
        f1.y = Alds[(16 + mrow) * n + kb * 4 + ksel + 1];
        acc00 = __builtin_amdgcn_wmma_f32_16x16x4_f32(false, f0, false, f0, (short)0, acc00, false, false);
        acc01 = __builtin_amdgcn_wmma_f32_16x16x4_f32(false, f0, false, f1, (short)0, acc01, false, false);
        acc10 = __builtin_amdgcn_wmma_f32_16x16x4_f32(false, f1, false, f0, (short)0, acc10, false, false);
        acc11 = __builtin_amdgcn_wmma_f32_16x16x4_f32(false, f1, false, f1, (short)0, acc11, false, false);
    }

    // Scatter C tiles: 2*G (diagonal +1 added after), right half zeroed.
    const int rofs = (lane < 16) ? 0 : 8;
    #pragma unroll
    for (int v = 0; v < 8; ++v) {
        int r0 = v + rofs, r1 = 16 + v + rofs;
        GJ[r0 * 64 + mrow]      = 2.0f * acc00[v];
        GJ[r0 * 64 + 16 + mrow] = 2.0f * acc01[v];
        GJ[r1 * 64 + mrow]      = 2.0f * acc10[v];
        GJ[r1 * 64 + 16 + mrow] = 2.0f * acc11[v];
    }
    {
        float4 z4 = make_float4(0.f, 0.f, 0.f, 0.f);
        #pragma unroll
        for (int c = 0; c < 32; c += 4) *(float4*)&GJ[lane * 64 + 32 + c] = z4;
    }
    wave_sync();
    GJ[lane * 64 + lane]      += 1.0f;   // G = 2AA^T + I
    GJ[lane * 64 + 32 + lane]  = 1.0f;   // augmented identity
    wave_sync();

    // ---- Gauss-Jordan inverse of SPD G (no pivoting; eig(G) >= 1) ----
    #pragma unroll 1
    for (int k = 0; k < 32; ++k) {
        float pinv = 1.0f / GJ[k * 64 + k];
        float fct  = GJ[lane * 64 + k] * pinv;
        bool  isPiv = (lane == k);
        #pragma unroll 8
        for (int c = 0; c < 64; ++c) {
            float rkc  = GJ[k * 64 + c];       // uniform read, pre-store in program order
            float mine = GJ[lane * 64 + c];
            GJ[lane * 64 + c] = isPiv ? rkc * pinv : mine - fct * rkc;
        }
        wave_sync();
    }

    // lane l keeps row l of G^-1 in VGPRs
    float Gr[32];
    #pragma unroll
    for (int i = 0; i < 32; ++i) Gr[i] = GJ[lane * 64 + 32 + i];

    const float bb  = b[(size_t)bidx * 32 + lane];
    const float sgn = (lane < 16) ? 1.0f : -1.0f;
    const float un  = u_nom[(size_t)bidx * 16 + (lane & 15)] * sgn;
    const int   max_iter = *max_iter_p;

    // ---- Davis-Yin fixed-point iteration ----
    // z (64) split: zlo = z[lane], zhi = z[32+lane]
    float zlo = 0.0f, zhi = 0.0f;

    auto step = [&](float& resOut) {
        float xlo = fmaxf(zlo, 0.0f), xhi = fmaxf(zhi, 0.0f);
        // grad[l] = sgn*(u - u_nom)[l&15] for l<32, 0 for upper half
        float s   = xlo - __shfl_xor(xlo, 16, 32);
        float g   = s - un;
        float vlo = 2.0f * xlo - zlo - DYS_ALPHA * g;
        float vhi = 2.0f * xhi - zhi;
        // d = v[0:16] - v[16:32]
        float dd  = vlo - __shfl_xor(vlo, 16, 32);
        if (lane < 16) bcd[lane] = dd;
        wave_sync();
        // r = A d + v3 - b           (32x16 matvec, row per lane)
        float r = vhi - bb;
        #pragma unroll
        for (int j = 0; j < 16; ++j) r = fmaf(Ar[j], bcd[j], r);
        bcr[lane] = r;
        wave_sync();
        // w = G^-1 r                 (32x32 matvec, row per lane)
        float w = 0.0f;
        #pragma unroll
        for (int i = 0; i < 32; ++i) w = fmaf(Gr[i], bcr[i], w);
        bcw[lane] = w;
        wave_sync();
        // t = A^T w                  (16x32 matvec; column cached in VGPRs)
        float t = 0.0f;
        #pragma unroll
        for (int i = 0; i < 32; ++i) t = fmaf(Atcol[i], bcw[i], t);
        // y = v - [t; -t; w],  z_new = z + y - x
        float nzlo = zlo + (vlo - sgn * t) - xlo;
        float nzhi = zhi + (vhi - w)       - xhi;
        float diff = fmaxf(fabsf(nzlo - zlo), fabsf(nzhi - zhi));
        #pragma unroll
        for (int off = 16; off > 0; off >>= 1)
            diff = fmaxf(diff, __shfl_xor(diff, off, 32));
        zlo = nzlo; zhi = nzhi;
        resOut = diff;
        wave_sync();
    };

    float res = __builtin_inff();
    int it = 0;
    #pragma unroll 1
    while (res >= DYS_TOL && it < max_iter - 1) { step(res); ++it; }
    float dummy; step(dummy);                 // final T step (n_grad_iters=1)

    // ---- outputs: u* = z* M^T,  z* ----
    float u = zlo - __shfl_xor(zlo, 16, 32);
    if (lane < 16) out_u[(size_t)bidx * 16 + lane] = u;
    out_z[(size_t)bidx * 64 + lane]      = zlo;
    out_z[(size_t)bidx * 64 + 32 + lane] = zhi;
}

extern "C" void kernel_launch(void* const* d_in, const int* in_sizes, int n_in,
                              void* d_out, int out_size, void* d_ws, size_t ws_size,
                              hipStream_t stream) {
    (void)n_in; (void)out_size; (void)d_ws; (void)ws_size;
    const float* u_nom = (const float*)d_in[0];
    const float* A     = (const float*)d_in[1];
    const float* b     = (const float*)d_in[2];
    const int*   mi    = (const int*)d_in[3];
    const int batch = in_sizes[0] / 16;     // u_nom is (B,16)
    float* out_u = (float*)d_out;
    float* out_z = out_u + (size_t)batch * 16;
    dys_projector_kernel<<<batch, 32, 0, stream>>>(u_nom, A, b, mi, out_u, out_z);
}